// FusionLoss_3332894622339
// MI455X (gfx1250) — compile-verified
//
#include <hip/hip_runtime.h>
#include <math.h>

#define B_    16
#define H_    512
#define W_    512
#define HW_   (H_*W_)
#define CHW_  (3*HW_)

#define SEG_OUT 60
#define NSEG    9
#define ROWS    16
#define NBAND   (H_/ROWS)          // 32
#define NPART   (NSEG*NBAND*B_)    // 4608 = 72*64

typedef __attribute__((ext_vector_type(2))) float v2f;
typedef __attribute__((ext_vector_type(8))) float v8f;

__device__ __forceinline__ float clamp01(float x) { return fminf(fmaxf(x, 0.f), 1.f); }

// butterfly reduction: ds_swizzle xor path, no clamp overhead
__device__ __forceinline__ float wred(float x) {
#pragma unroll
  for (int m = 16; m; m >>= 1) x += __shfl_xor(x, m, 32);
  return x;
}

// raw ds_bpermute with precomputed byte address (1 DS op, no VALU setup)
__device__ __forceinline__ float lane_gather(int addr, float v) {
  return __int_as_float(__builtin_amdgcn_ds_bpermute(addr, __float_as_int(v)));
}

// ---- Stage 1: per-(batch,chunk) partial sums of gray(ir) -------------------
__global__ __launch_bounds__(256) void k_irsum(const float* __restrict__ ir,
                                               float* __restrict__ irpart) {
  __shared__ float red[256];
  const int b = blockIdx.x >> 4, chunk = blockIdx.x & 15;
  const size_t base = (size_t)b * CHW_ + (size_t)chunk * 16384 + threadIdx.x;
  float s = 0.f;
#pragma unroll 4
  for (int k = 0; k < 64; ++k) {
    const size_t p = base + (size_t)k * 256;
    s += 0.2989f * ir[p] + 0.587f * ir[p + HW_] + 0.114f * ir[p + 2 * HW_];
  }
  red[threadIdx.x] = s;
  __syncthreads();
  for (int o = 128; o; o >>= 1) {
    if (threadIdx.x < (unsigned)o) red[threadIdx.x] += red[threadIdx.x + o];
    __syncthreads();
  }
  if (threadIdx.x == 0) irpart[blockIdx.x] = red[0];
}

// ---- Stage 1b: reduce 256 partials -> 16 per-batch means (async->LDS path) -
__global__ __launch_bounds__(256) void k_irmean(const float* __restrict__ irpart,
                                                float* __restrict__ irmean) {
  __shared__ float lds[256];
  const unsigned loff = (unsigned)(size_t)(&lds[threadIdx.x]);
  const unsigned long long g = (unsigned long long)(size_t)(irpart + threadIdx.x);
  // CDNA5 async global->LDS DMA (ASYNCcnt), then wait.
  asm volatile("global_load_async_to_lds_b32 %0, %1, off" :: "v"(loff), "v"(g) : "memory");
  asm volatile("s_wait_asynccnt 0" ::: "memory");
  __syncthreads();
  if (threadIdx.x < 16) {
    float s = 0.f;
#pragma unroll
    for (int k = 0; k < 16; ++k) s += lds[threadIdx.x * 16 + k];
    irmean[threadIdx.x] = s * (1.0f / (float)HW_);
  }
}

// ---- Stage 2: fused transform + separable Sobel + L1 partials --------------
__global__ __launch_bounds__(32) void k_main(const float* __restrict__ visp,
                                             const float* __restrict__ irp,
                                             const float* __restrict__ outp,
                                             const float* __restrict__ maskp,
                                             const float* __restrict__ irmean,
                                             float* __restrict__ partials) {
  const int lane = threadIdx.x;                 // 0..31, wave32
  const int s = blockIdx.x;                     // column segment
  const int band = blockIdx.y;                  // row band
  const int b = blockIdx.z;                     // batch
  const int olo = s * SEG_OUT;
  const int ohi = min(olo + SEG_OUT, W_);
  const int col = olo - 2 + 2 * lane;           // even -> 8B-aligned float2 loads
  const bool inw = (col >= 0) && (col < W_);
  const int r0 = band * ROWS;
  const float nIr = -0.8f * irmean[b];          // (1-1.8)*mean

  const bool outc[2] = { inw && (col     >= olo) && (col     < ohi),
                         inw && (col + 1 >= olo) && (col + 1 < ohi) };

  // static neighbor-exchange addresses for ds_bpermute (bytes, bits[6:2] used).
  // lane 0 / lane 31 wrap reads are never consumed (outc masks them off).
  const int addrL = (lane - 1) * 4;             // value of lane-1
  const int addrR = (lane + 1) * 4;             // value of lane+1

  // rolling 3-row rings of separable H-pass, per component(2) per field(9)
  float hd[2][9][3], hs[2][9][3];
  float accCon = 0.f, accGx = 0.f, accGy = 0.f, accCb = 0.f, accCr = 0.f;

#pragma unroll
  for (int rr = -1; rr <= ROWS; ++rr) {
    const int r = r0 + rr;
    float t[2][9];
#pragma unroll
    for (int k = 0; k < 2; ++k)
#pragma unroll
      for (int f = 0; f < 9; ++f) t[k][f] = 0.f;   // zero 'SAME' padding

    if (inw && (r >= 0) && (r < H_)) {
      const size_t rb = (size_t)b * CHW_ + (size_t)r * W_ + (size_t)col;
#pragma unroll
      for (int c = 0; c < 3; ++c) {
        const size_t ix = rb + (size_t)c * HW_;
        const float2 o = *(const float2*)(outp + ix);
        const float2 m = *(const float2*)(maskp + ix);
        const float2 v = *(const float2*)(visp + ix);
        const float2 q = *(const float2*)(irp + ix);
        t[0][c]     = o.x * m.x;                       // Fuse
        t[1][c]     = o.y * m.y;
        t[0][3 + c] = clamp01(sqrtf(v.x));             // R_vis = clip(vis^0.5)
        t[1][3 + c] = clamp01(sqrtf(v.y));
        t[0][6 + c] = clamp01(fmaf(1.8f, q.x, nIr));   // R_ir
        t[1][6 + c] = clamp01(fmaf(1.8f, q.y, nIr));
      }
      if (r + 8 < H_) {                                // gfx1250 global_prefetch
        const size_t pf = rb + (size_t)(8 * W_);
        __builtin_prefetch(outp + pf, 0, 0);
        __builtin_prefetch(maskp + pf, 0, 0);
        __builtin_prefetch(visp + pf, 0, 0);
        __builtin_prefetch(irp + pf, 0, 0);
      }
    }

    // horizontal pass via precomputed ds_bpermute (no LDS memory, no clamps)
    const int slot = (rr + 1) % 3;
#pragma unroll
    for (int f = 0; f < 9; ++f) {
      const float lft = lane_gather(addrL, t[1][f]);   // col-1
      const float rgt = lane_gather(addrR, t[0][f]);   // col+2
      hd[0][f][slot] = t[1][f] - lft;
      hd[1][f][slot] = rgt - t[0][f];
      hs[0][f][slot] = lft + 2.f * t[0][f] + t[1][f];
      hs[1][f][slot] = t[0][f] + 2.f * t[1][f] + rgt;
    }

    // pointwise losses at row r
    if (rr >= 0 && rr < ROWS) {
#pragma unroll
      for (int k = 0; k < 2; ++k) {
        if (!outc[k]) continue;
        const float* tf = t[k];
        const float YF = 0.299f * tf[0] + 0.587f * tf[1] + 0.114f * tf[2];
        const float YV = 0.299f * tf[3] + 0.587f * tf[4] + 0.114f * tf[5];
        accCr += fabsf((tf[0] - YF) * 0.713f - (tf[3] - YV) * 0.713f);
        accCb += fabsf((tf[2] - YF) * 0.564f - (tf[5] - YV) * 0.564f);
        accCon += fabsf(tf[0] - fmaxf(tf[3], tf[6]))
                + fabsf(tf[1] - fmaxf(tf[4], tf[7]))
                + fabsf(tf[2] - fmaxf(tf[5], tf[8]));
      }
    }

    // Sobel output for row r-1 (vertical pass over the ring)
    if (rr >= 1) {
      const int sT = (rr - 1) % 3, sM = rr % 3, sB = (rr + 1) % 3;
#pragma unroll
      for (int k = 0; k < 2; ++k) {
        if (!outc[k]) continue;
#pragma unroll
        for (int c = 0; c < 3; ++c) {
          const float gxF = hd[k][c][sT]     + 2.f * hd[k][c][sM]     + hd[k][c][sB];
          const float gxV = hd[k][3 + c][sT] + 2.f * hd[k][3 + c][sM] + hd[k][3 + c][sB];
          const float gxI = hd[k][6 + c][sT] + 2.f * hd[k][6 + c][sM] + hd[k][6 + c][sB];
          accGx += fabsf(gxF - fmaxf(gxV, gxI));
          const float gyF = hs[k][c][sT]     - hs[k][c][sB];
          const float gyV = hs[k][3 + c][sT] - hs[k][3 + c][sB];
          const float gyI = hs[k][6 + c][sT] - hs[k][6 + c][sB];
          accGy += fabsf(gyF - fmaxf(gyV, gyI));
        }
      }
    }
  }

  accCon = wred(accCon); accGx = wred(accGx); accGy = wred(accGy);
  accCb  = wred(accCb);  accCr = wred(accCr);
  if (lane == 0) {
    const float wCon = 0.5f / 12582912.f;   // 0.5 * 1/(B*C*H*W)
    const float wG   = 0.1f / 12582912.f;   // 0.2 * 0.5 * 1/N
    const float wCol = 1.0f / 4194304.f;    // 1.0 * 1/(B*H*W)
    const int flat = (blockIdx.z * gridDim.y + blockIdx.y) * gridDim.x + blockIdx.x;
    partials[flat] = wCon * accCon + wG * (accGx + accGy) + wCol * (accCb + accCr);
  }
}

// ---- Stage 3: grand sum of 4608 weighted partials via WMMA (ones x data) ---
__global__ __launch_bounds__(32) void k_final(const float* __restrict__ partials,
                                              float* __restrict__ out) {
#if __has_builtin(__builtin_amdgcn_wmma_f32_16x16x4_f32)
  v2f a; a[0] = 1.f; a[1] = 1.f;            // A = all-ones 16x4
  v8f acc;
#pragma unroll
  for (int i = 0; i < 8; ++i) acc[i] = 0.f;
  for (int i = 0; i < NPART / 64; ++i) {    // 72 chunks of 64 partials
    v2f bv;
    bv[0] = partials[i * 64 + threadIdx.x];
    bv[1] = partials[i * 64 + 32 + threadIdx.x];
    // D[m][n] = sum_k B[k][n] + C[m][n]; every slot is one partial.
    acc = __builtin_amdgcn_wmma_f32_16x16x4_f32(false, a, false, bv,
                                                (short)0, acc, false, false);
  }
  // acc[0]: lanes 0..15 hold row M=0 col-sums, lanes 16..31 row M=8 (identical)
  float v = acc[0];
  v += __shfl_xor(v, 16, 32); v += __shfl_xor(v, 8, 32);
  v += __shfl_xor(v, 4, 32);  v += __shfl_xor(v, 2, 32);
  v += __shfl_xor(v, 1, 32);
  if (threadIdx.x == 0) out[0] = 0.5f * v;  // rows duplicated -> halve
#else
  if (threadIdx.x == 0) {
    float s = 0.f;
    for (int i = 0; i < NPART; ++i) s += partials[i];
    out[0] = s;
  }
#endif
}

extern "C" void kernel_launch(void* const* d_in, const int* in_sizes, int n_in,
                              void* d_out, int out_size, void* d_ws, size_t ws_size,
                              hipStream_t stream) {
  (void)in_sizes; (void)n_in; (void)out_size; (void)ws_size;
  const float* vis  = (const float*)d_in[0];
  const float* ir   = (const float*)d_in[1];
  const float* outp = (const float*)d_in[2];
  const float* mask = (const float*)d_in[3];
  float* ws      = (float*)d_ws;
  float* irpart  = ws;          // 256 floats
  float* irmean  = ws + 256;    // 16 floats
  float* partials= ws + 272;    // 4608 floats

  k_irsum <<<256, 256, 0, stream>>>(ir, irpart);
  k_irmean<<<1,   256, 0, stream>>>(irpart, irmean);
  k_main  <<<dim3(NSEG, NBAND, B_), 32, 0, stream>>>(vis, ir, outp, mask, irmean, partials);
  k_final <<<1, 32, 0, stream>>>(partials, (float*)d_out);
}